// FasterRCNN_28913719837243
// MI455X (gfx1250) — compile-verified
//
#include <hip/hip_runtime.h>

#define FH 25
#define FW 38
#define CCH 2048
#define NBOX 2000
#define NKEEP 8
#define KFC1 100352
#define NFC 4096
#define NOUT 405

typedef __attribute__((ext_vector_type(2))) float v2f;
typedef __attribute__((ext_vector_type(8))) float v8f;

__global__ void zero_f(float* __restrict__ p, int n) {
  int i = blockIdx.x * blockDim.x + threadIdx.x;
  if (i < n) p[i] = 0.0f;
}

// ---------------- NMS + box-param kernel (1 block, 256 threads) ----------------
__global__ void nms_kernel(const float* __restrict__ rois, const float* __restrict__ scores,
                           int* __restrict__ roi_out) {
  __shared__ float m[NBOX];
  __shared__ float redV[256];
  __shared__ int   redI[256];
  __shared__ int   ssel[NKEEP];
  __shared__ float cur[4];
  const int tid = threadIdx.x;
  const float NEG = -__builtin_inff();

  for (int i = tid; i < NBOX; i += 256) {
    float x1 = rois[4*i], y1 = rois[4*i+1], x2 = rois[4*i+2], y2 = rois[4*i+3];
    bool valid = (x1 >= 0.f) && (y1 >= 0.f) && (x2 <= 1216.f) && (y2 <= 800.f);
    m[i] = valid ? scores[2*i+1] : NEG;
  }
  __syncthreads();

  for (int it = 0; it < NKEEP; ++it) {
    float bv = NEG; int bi = 0;
    for (int i = tid; i < NBOX; i += 256) {
      float v = m[i];
      if (v > bv) { bv = v; bi = i; }
    }
    redV[tid] = bv; redI[tid] = bi;
    __syncthreads();
    for (int s = 128; s > 0; s >>= 1) {
      if (tid < s) {
        float ov = redV[tid+s]; int oi = redI[tid+s];
        if (ov > redV[tid] || (ov == redV[tid] && oi < redI[tid])) {
          redV[tid] = ov; redI[tid] = oi;
        }
      }
      __syncthreads();
    }
    if (tid == 0) {
      int s = redI[0];
      ssel[it] = s;
      cur[0] = rois[4*s]; cur[1] = rois[4*s+1]; cur[2] = rois[4*s+2]; cur[3] = rois[4*s+3];
    }
    __syncthreads();
    float bx1 = cur[0], by1 = cur[1], bx2 = cur[2], by2 = cur[3];
    float a1 = (bx2 - bx1) * (by2 - by1);
    for (int i = tid; i < NBOX; i += 256) {
      float x1 = rois[4*i], y1 = rois[4*i+1], x2 = rois[4*i+2], y2 = rois[4*i+3];
      float ix1 = fmaxf(bx1, x1), iy1 = fmaxf(by1, y1);
      float ix2 = fminf(bx2, x2), iy2 = fminf(by2, y2);
      float inter = fmaxf(ix2 - ix1, 0.f) * fmaxf(iy2 - iy1, 0.f);
      float a2 = (x2 - x1) * (y2 - y1);
      float iou = inter / (a1 + a2 - inter);
      if (iou > 0.6f) m[i] = NEG;
    }
    __syncthreads();
  }

  if (tid < NKEEP) {
    int s = ssel[tid];
    float b0  = rois[4*s]   * (1.f/32.f);
    float b1v = rois[4*s+1] * (1.f/32.f);
    float b2v = rois[4*s+2] * (1.f/32.f);
    float b3v = rois[4*s+3] * (1.f/32.f);
    int x = (int)floorf(b0 + 0.5f);
    int y = (int)floorf(b1v + 0.5f);
    int w = (int)floorf(b2v + 1.0f);
    int h = (int)floorf(b3v + 1.0f);
    bool bad = (x < 0) || (x >= FW) || (y < 0) || (y >= FH) ||
               (w < 1) || (x + w > FW) || (h < 1) || (y + h > FH);
    if (bad) { x = 0; y = 0; w = FW; h = FH; }
    roi_out[4*tid]   = x; roi_out[4*tid+1] = y;
    roi_out[4*tid+2] = w; roi_out[4*tid+3] = h;
  }
}

// ---------------- adaptive 7x7 max-pool, one block per (roi, channel) ----------------
__global__ void roipool_kernel(const float* __restrict__ feat, const int* __restrict__ roi,
                               float* __restrict__ flat) {
  const int r = blockIdx.x;   // 0..7
  const int c = blockIdx.y;   // 0..2047
  const int t = threadIdx.x;  // 64 threads, 49 active
  if (t >= 49) return;
  const int x = roi[4*r], y = roi[4*r+1], w = roi[4*r+2], h = roi[4*r+3];
  const int i = t / 7, j = t % 7;
  int r0 = y + (i * h) / 7;
  int r1 = y + ((i + 1) * h + 6) / 7;
  int c0 = x + (j * w) / 7;
  int c1 = x + ((j + 1) * w + 6) / 7;
  if (r0 < 0) r0 = 0;  if (r1 > FH) r1 = FH;
  if (c0 < 0) c0 = 0;  if (c1 > FW) c1 = FW;
  float mx = -__builtin_inff();
  const float* f = feat + (size_t)c * (FH * FW);
  for (int rr = r0; rr < r1; ++rr)
    for (int cc = c0; cc < c1; ++cc)
      mx = fmaxf(mx, f[rr * FW + cc]);
  flat[(size_t)r * KFC1 + c * 49 + t] = mx;
}

// ---------------- out(16xN, stride 4096) += A(16xK) * W(NxK)^T via f32 WMMA ----------------
// grid.x = N/16 (N tile), grid.y = K/(8*KPW); block = 256 threads (8 waves).
// KPW = K elements handled per wave (compile-time, multiple of 16).
template <int KPW>
__global__ void gemm_wmma_f32(const float* __restrict__ A, const float* __restrict__ W,
                              float* __restrict__ out, int K) {
  const int n0   = blockIdx.x * 16;
  const int wave = threadIdx.x >> 5;
  const int lane = threadIdx.x & 31;
  const int k0   = (blockIdx.y * 8 + wave) * KPW;
  const int row  = lane & 15;                // A: M=row ; B: N=row
  const int kh   = (lane >> 4) << 1;         // K-half: lanes 0-15 -> K0/1, 16-31 -> K2/3
  const float* aPtr = A + (size_t)row * K + k0 + kh;
  const float* wPtr = W + (size_t)(n0 + row) * K + k0 + kh;

  v8f acc0 = {};
  v8f acc1 = {};
  #pragma unroll 1
  for (int kk = 0; kk < KPW; kk += 16) {
    // 8 loads issued together, then 4 WMMAs on 2 interleaved accumulators
    v2f a0 = *(const v2f*)(aPtr + kk);
    v2f a1 = *(const v2f*)(aPtr + kk + 4);
    v2f a2 = *(const v2f*)(aPtr + kk + 8);
    v2f a3 = *(const v2f*)(aPtr + kk + 12);
    v2f b0 = __builtin_nontemporal_load((const v2f*)(wPtr + kk));
    v2f b1 = __builtin_nontemporal_load((const v2f*)(wPtr + kk + 4));
    v2f b2 = __builtin_nontemporal_load((const v2f*)(wPtr + kk + 8));
    v2f b3 = __builtin_nontemporal_load((const v2f*)(wPtr + kk + 12));
    acc0 = __builtin_amdgcn_wmma_f32_16x16x4_f32(false, a0, false, b0, (short)0, acc0, false, false);
    acc1 = __builtin_amdgcn_wmma_f32_16x16x4_f32(false, a1, false, b1, (short)0, acc1, false, false);
    acc0 = __builtin_amdgcn_wmma_f32_16x16x4_f32(false, a2, false, b2, (short)0, acc0, false, false);
    acc1 = __builtin_amdgcn_wmma_f32_16x16x4_f32(false, a3, false, b3, (short)0, acc1, false, false);
  }
  v8f acc = acc0 + acc1;

  __shared__ float red[8][32][8];
  #pragma unroll
  for (int v = 0; v < 8; ++v) red[wave][lane][v] = acc[v];
  __syncthreads();
  if (wave == 0) {
    #pragma unroll
    for (int v = 0; v < 8; ++v) {
      float s = 0.f;
      #pragma unroll
      for (int wv = 0; wv < 8; ++wv) s += red[wv][lane][v];
      int mrow = v + ((lane >> 4) << 3);     // C/D layout: lanes 16-31 hold M=8..15
      atomicAdd(&out[(size_t)mrow * NFC + n0 + (lane & 15)], s);
    }
  }
}

// ---------------- add bias and re-pad rows 8..15 with zeros ----------------
__global__ void bias_pad_kernel(const float* __restrict__ acc, const float* __restrict__ bias,
                                float* __restrict__ outp) {
  int idx = blockIdx.x * blockDim.x + threadIdx.x;   // 16*4096
  int r = idx >> 12, j = idx & 4095;
  outp[idx] = (r < NKEEP) ? acc[idx] + bias[j] : 0.0f;
}

// ---------------- classification + localization heads, one wave per output element ----------------
__global__ void heads_kernel(const float* __restrict__ fc2acc, const float* __restrict__ b2,
                             const float* __restrict__ wc, const float* __restrict__ bc,
                             const float* __restrict__ wl, const float* __restrict__ bl,
                             float* __restrict__ outp) {
  int wave = threadIdx.x >> 5, lane = threadIdx.x & 31;
  int e = blockIdx.x * 8 + wave;             // 0..3239
  int roi = e / NOUT, col = e % NOUT;
  const float* o = fc2acc + (size_t)roi * NFC;
  const float* wr; float bias;
  if (col < 81) { wr = wc + (size_t)col * NFC; bias = bc[col]; }
  else          { wr = wl + (size_t)(col - 81) * NFC; bias = bl[col - 81]; }
  float s = 0.f;
  for (int k = lane; k < NFC; k += 32) s += (o[k] + b2[k]) * wr[k];
  #pragma unroll
  for (int off = 16; off; off >>= 1) s += __shfl_down(s, off);
  if (lane == 0) outp[(size_t)roi * NOUT + col] = s + bias;
}

extern "C" void kernel_launch(void* const* d_in, const int* in_sizes, int n_in,
                              void* d_out, int out_size, void* d_ws, size_t ws_size,
                              hipStream_t stream) {
  const float* feat   = (const float*)d_in[0];
  const float* rois   = (const float*)d_in[1];
  const float* scores = (const float*)d_in[2];
  const float* w1     = (const float*)d_in[3];
  const float* b1     = (const float*)d_in[4];
  const float* w2     = (const float*)d_in[5];
  const float* b2     = (const float*)d_in[6];
  const float* wc     = (const float*)d_in[7];
  const float* bc     = (const float*)d_in[8];
  const float* wl     = (const float*)d_in[9];
  const float* bl     = (const float*)d_in[10];
  float* out = (float*)d_out;

  float* flat   = (float*)d_ws;              // [16][100352] (rows 8..15 zero)
  float* fc1acc = flat + (size_t)16 * KFC1;  // [16][4096]
  float* fc1f   = fc1acc + 16 * NFC;         // [16][4096]
  float* fc2acc = fc1f + 16 * NFC;           // [16][4096]
  int*   roi_i  = (int*)(fc2acc + 16 * NFC); // [8][4]

  zero_f<<<(8 * KFC1 + 255) / 256, 256, 0, stream>>>(flat + (size_t)8 * KFC1, 8 * KFC1);
  zero_f<<<(16 * NFC + 255) / 256, 256, 0, stream>>>(fc1acc, 16 * NFC);
  zero_f<<<(16 * NFC + 255) / 256, 256, 0, stream>>>(fc2acc, 16 * NFC);

  nms_kernel<<<1, 256, 0, stream>>>(rois, scores, roi_i);
  roipool_kernel<<<dim3(NKEEP, CCH), 64, 0, stream>>>(feat, roi_i, flat);

  // fc1: K=100352 = 49 chunks * 8 waves * 256 K/wave -> 64 WMMAs per wave
  gemm_wmma_f32<256><<<dim3(NFC / 16, 49), 256, 0, stream>>>(flat, w1, fc1acc, KFC1);
  bias_pad_kernel<<<(16 * NFC) / 256, 256, 0, stream>>>(fc1acc, b1, fc1f);

  // fc2: K=4096 = 16 chunks * 8 waves * 32 K/wave -> 8 WMMAs per wave
  gemm_wmma_f32<32><<<dim3(NFC / 16, 16), 256, 0, stream>>>(fc1f, w2, fc2acc, NFC);

  heads_kernel<<<405, 256, 0, stream>>>(fc2acc, b2, wc, bc, wl, bl, out);
}